// GraphAtt2Mesh_59889023975747
// MI455X (gfx1250) — compile-verified
//
#include <hip/hip_runtime.h>
#include <hip/hip_bf16.h>

#define NN 100000      // nodes
#define NE 3200000     // edges (without self loops)
#define NG 64          // graphs
#define HEADS 8

typedef __attribute__((ext_vector_type(2))) float v2f;
typedef __attribute__((ext_vector_type(8))) float v8f;

// ---- order-preserving float <-> uint encoding for atomic max ----
__device__ __forceinline__ unsigned int enc_f32(float f) {
  unsigned int u = __float_as_uint(f);
  return (u & 0x80000000u) ? ~u : (u | 0x80000000u);
}
__device__ __forceinline__ float dec_f32(unsigned int e) {
  return (e & 0x80000000u) ? __uint_as_float(e ^ 0x80000000u)
                           : __uint_as_float(~e);
}

// ================= WMMA f32 GEMM: C[M,F] = A[M,K] * W[K,F] (+bias, relu) ====
// one wave -> two 16x16 tiles of C (32 rows x 16 cols): each B fragment is
// reused by two back-to-back v_wmma_f32_16x16x4_f32, halving W-load traffic.
// Requires M % 32 == 0 (true for 100000 and 64).
__global__ void wmma_gemm_f32(const float* __restrict__ A,
                              const float* __restrict__ W,
                              const float* __restrict__ bias,
                              float* __restrict__ C,
                              int M, int K, int F, int relu) {
  int m0 = blockIdx.x * 32;
  int ntile = blockIdx.y * blockDim.y + threadIdx.y;
  int n0 = ntile * 16;
  if (n0 >= F) return;                 // wave-uniform exit (EXEC stays full)
  int lane = threadIdx.x;
  int half = lane >> 4;                // 0: K pair {0,1}, 1: K pair {2,3}
  int lm   = lane & 15;

  const float* Arow0 = A + (size_t)(m0 + lm) * K + 2 * half;
  const float* Arow1 = A + (size_t)(m0 + 16 + lm) * K + 2 * half;
  __builtin_prefetch(Arow0, 0, 1);     // global_prefetch_b8
  __builtin_prefetch(Arow1, 0, 1);

  int ncol = n0 + lm;
  int nc = ncol < F ? ncol : F - 1;    // clamp keeps loads in-bounds
  float nmask = ncol < F ? 1.0f : 0.0f;

  v8f acc0 = {}, acc1 = {};
  for (int k0 = 0; k0 < K; k0 += 4) {
    v2f a0, a1, b;
    int kb = k0 + 2 * half;
    b.x = W[(size_t)kb * F + nc] * nmask;
    b.y = W[(size_t)(kb + 1) * F + nc] * nmask;
    a0.x = Arow0[k0]; a0.y = Arow0[k0 + 1];
    a1.x = Arow1[k0]; a1.y = Arow1[k0 + 1];
    acc0 = __builtin_amdgcn_wmma_f32_16x16x4_f32(false, a0, false, b,
                                                 (short)0, acc0, false, false);
    acc1 = __builtin_amdgcn_wmma_f32_16x16x4_f32(false, a1, false, b,
                                                 (short)0, acc1, false, false);
  }
  if (ncol < F) {
    float bv = bias ? bias[ncol] : 0.0f;
#pragma unroll
    for (int r = 0; r < 8; ++r) {      // D layout: VGPR r -> M = r (+8 hi half)
      int m = m0 + half * 8 + r;
      float v0 = acc0[r] + bv;
      float v1 = acc1[r] + bv;
      if (relu) { v0 = fmaxf(v0, 0.0f); v1 = fmaxf(v1, 0.0f); }
      C[(size_t)m * F + ncol] = v0;
      C[(size_t)(m + 16) * F + ncol] = v1;
    }
  }
}

// ====== per-node attention logits: asrc[n,h] = <h[n,h,:], a_src[h,:]> =======
template <int C>
__global__ void node_alpha_kernel(const float* __restrict__ H,
                                  const float* __restrict__ avs,
                                  const float* __restrict__ avd,
                                  float* __restrict__ asrc,
                                  float* __restrict__ adst) {
  int i = blockIdx.x * blockDim.x + threadIdx.x;
  if (i >= NN * HEADS) return;
  int node = i >> 3, h = i & 7;
  const float4* row = (const float4*)(H + (size_t)node * HEADS * C + h * C);
  const float4* vs  = (const float4*)(avs + h * C);
  const float4* vd  = (const float4*)(avd + h * C);
  float s1 = 0.f, s2 = 0.f;
#pragma unroll
  for (int c = 0; c < C / 4; ++c) {
    float4 v = row[c], w1 = vs[c], w2 = vd[c];
    s1 += v.x * w1.x + v.y * w1.y + v.z * w1.z + v.w * w1.w;
    s2 += v.x * w2.x + v.y * w2.y + v.z * w2.z + v.w * w2.w;
  }
  asrc[i] = s1;
  adst[i] = s2;
}

__device__ __forceinline__ void edge_sd(const int* __restrict__ ei, int e,
                                        int& s, int& d) {
  if (e < NE) { s = ei[e]; d = ei[NE + e]; }
  else        { s = e - NE; d = s; }       // appended self-loops
}

// ====== pass 1: segment max over dst (encoded-uint atomic max) ==============
__global__ void edge_max_kernel(const int* __restrict__ ei,
                                const float* __restrict__ asrc,
                                const float* __restrict__ adst,
                                unsigned int* __restrict__ menc) {
  int e = blockIdx.x * blockDim.x + threadIdx.x;
  if (e >= NE + NN) return;
  int s, d; edge_sd(ei, e, s, d);
#pragma unroll
  for (int h = 0; h < HEADS; ++h) {
    float v = asrc[s * HEADS + h] + adst[d * HEADS + h];
    v = v > 0.f ? v : 0.2f * v;              // leaky relu
    atomicMax(&menc[d * HEADS + h], enc_f32(v));
  }
}

// ====== pass 2: segment sum of exp(v - max) =================================
__global__ void edge_sum_kernel(const int* __restrict__ ei,
                                const float* __restrict__ asrc,
                                const float* __restrict__ adst,
                                const unsigned int* __restrict__ menc,
                                float* __restrict__ den) {
  int e = blockIdx.x * blockDim.x + threadIdx.x;
  if (e >= NE + NN) return;
  int s, d; edge_sd(ei, e, s, d);
#pragma unroll
  for (int h = 0; h < HEADS; ++h) {
    float v = asrc[s * HEADS + h] + adst[d * HEADS + h];
    v = v > 0.f ? v : 0.2f * v;
    atomicAdd(&den[d * HEADS + h], __expf(v - dec_f32(menc[d * HEADS + h])));
  }
}

// ====== pass 3: weighted scatter of messages; one thread per (edge, head) ===
template <int C>
__global__ void edge_scatter_kernel(const int* __restrict__ ei,
                                    const float* __restrict__ asrc,
                                    const float* __restrict__ adst,
                                    const unsigned int* __restrict__ menc,
                                    const float* __restrict__ den,
                                    const float* __restrict__ H,
                                    float* __restrict__ OUT) {
  long long i = (long long)blockIdx.x * blockDim.x + threadIdx.x;
  if (i >= (long long)(NE + NN) * HEADS) return;
  int h = (int)(i & 7);
  int e = (int)(i >> 3);
  int s, d; edge_sd(ei, e, s, d);
  float v = asrc[s * HEADS + h] + adst[d * HEADS + h];
  v = v > 0.f ? v : 0.2f * v;
  float alpha = __expf(v - dec_f32(menc[d * HEADS + h])) /
                (den[d * HEADS + h] + 1e-16f);
  const float4* hs4 = (const float4*)(H + (size_t)s * HEADS * C + h * C);
  float* od = OUT + (size_t)d * HEADS * C + h * C;
#pragma unroll
  for (int c4 = 0; c4 < C / 4; ++c4) {       // global_load_b128 gather
    float4 m = hs4[c4];
    atomicAdd(&od[c4 * 4 + 0], alpha * m.x);
    atomicAdd(&od[c4 * 4 + 1], alpha * m.y);
    atomicAdd(&od[c4 * 4 + 2], alpha * m.z);
    atomicAdd(&od[c4 * 4 + 3], alpha * m.w);
  }
}

__global__ void bias_relu_kernel(float* __restrict__ X,
                                 const float* __restrict__ b,
                                 long long n, int F) {
  long long i = (long long)blockIdx.x * blockDim.x + threadIdx.x;
  if (i >= n) return;
  X[i] = fmaxf(X[i] + b[(int)(i % F)], 0.0f);
}

// ====== global mean pool ====================================================
__global__ void pool_sum_kernel(const float* __restrict__ X,
                                const int* __restrict__ batch,
                                float* __restrict__ g,
                                float* __restrict__ cnt) {
  int i = blockIdx.x * blockDim.x + threadIdx.x;
  if (i >= NN * 128) return;
  int n = i >> 7, c = i & 127;
  int b = batch[n];
  atomicAdd(&g[b * 128 + c], X[i]);
  if (c == 0) atomicAdd(&cnt[b], 1.0f);
}
__global__ void pool_div_kernel(float* __restrict__ g,
                                const float* __restrict__ cnt) {
  int i = blockIdx.x * blockDim.x + threadIdx.x;
  if (i >= NG * 128) return;
  g[i] /= fmaxf(cnt[i >> 7], 1.0f);
}

static inline int cdiv(long long a, long long b) { return (int)((a + b - 1) / b); }

extern "C" void kernel_launch(void* const* d_in, const int* in_sizes, int n_in,
                              void* d_out, int out_size, void* d_ws, size_t ws_size,
                              hipStream_t stream) {
  const float* x      = (const float*)d_in[0];
  const int*   ei     = (const int*)d_in[1];
  const int*   batch  = (const int*)d_in[2];
  const float* W1     = (const float*)d_in[3];
  const float* a_src1 = (const float*)d_in[4];
  const float* a_dst1 = (const float*)d_in[5];
  const float* b1     = (const float*)d_in[6];
  const float* W2     = (const float*)d_in[7];
  const float* a_src2 = (const float*)d_in[8];
  const float* a_dst2 = (const float*)d_in[9];
  const float* b2     = (const float*)d_in[10];
  const float* fc1_w  = (const float*)d_in[11];
  const float* fc1_b  = (const float*)d_in[12];
  const float* fc2_w  = (const float*)d_in[13];
  const float* fc2_b  = (const float*)d_in[14];
  const float* fc3_w  = (const float*)d_in[15];
  const float* fc3_b  = (const float*)d_in[16];
  const float* fc4_w  = (const float*)d_in[17];
  const float* fc4_b  = (const float*)d_in[18];

  float* ws = (float*)d_ws;
  float* H1   = ws;                     // 6.4M f
  float* X2   = ws + 6400000;           // 6.4M f (layer-1 out, relu'd in place)
  float* H2   = ws + 12800000;          // 12.8M f
  float* OUT2 = ws;                     // 12.8M f, aliases H1+X2 (both dead by then)
  float* ASRC = ws + 25600000;          // 0.8M f
  float* ADST = ws + 26400000;          // 0.8M f
  unsigned int* MENC = (unsigned int*)(ws + 27200000);  // 0.8M u32
  float* DEN  = ws + 28000000;          // 0.8M f
  float* G    = ws + 28800000;          // 8192
  float* CNT  = ws + 28808192;          // 64
  float* G1   = ws + 28808256;          // 16384
  float* G2   = ws + 28824640;          // 32768
  float* G3   = ws + 28857408;          // 32768

  const int ET = NE + NN;
  dim3 gblk(32, 4);

  // ---------- GAT layer 1 (C=8, F=64) ----------
  wmma_gemm_f32<<<dim3(NN / 32, 1), gblk, 0, stream>>>(x, W1, nullptr, H1, NN, 128, 64, 0);
  node_alpha_kernel<8><<<cdiv((long long)NN * HEADS, 256), 256, 0, stream>>>(H1, a_src1, a_dst1, ASRC, ADST);
  hipMemsetAsync(MENC, 0, (size_t)NN * HEADS * 4, stream);
  hipMemsetAsync(DEN,  0, (size_t)NN * HEADS * 4, stream);
  hipMemsetAsync(X2,   0, (size_t)NN * 64 * 4, stream);
  edge_max_kernel<<<cdiv(ET, 256), 256, 0, stream>>>(ei, ASRC, ADST, MENC);
  edge_sum_kernel<<<cdiv(ET, 256), 256, 0, stream>>>(ei, ASRC, ADST, MENC, DEN);
  edge_scatter_kernel<8><<<cdiv((long long)ET * HEADS, 256), 256, 0, stream>>>(ei, ASRC, ADST, MENC, DEN, H1, X2);
  bias_relu_kernel<<<cdiv((long long)NN * 64, 256), 256, 0, stream>>>(X2, b1, (long long)NN * 64, 64);

  // ---------- GAT layer 2 (C=16, F=128) ----------
  wmma_gemm_f32<<<dim3(NN / 32, 2), gblk, 0, stream>>>(X2, W2, nullptr, H2, NN, 64, 128, 0);
  node_alpha_kernel<16><<<cdiv((long long)NN * HEADS, 256), 256, 0, stream>>>(H2, a_src2, a_dst2, ASRC, ADST);
  hipMemsetAsync(MENC, 0, (size_t)NN * HEADS * 4, stream);
  hipMemsetAsync(DEN,  0, (size_t)NN * HEADS * 4, stream);
  hipMemsetAsync(OUT2, 0, (size_t)NN * 128 * 4, stream);
  edge_max_kernel<<<cdiv(ET, 256), 256, 0, stream>>>(ei, ASRC, ADST, MENC);
  edge_sum_kernel<<<cdiv(ET, 256), 256, 0, stream>>>(ei, ASRC, ADST, MENC, DEN);
  edge_scatter_kernel<16><<<cdiv((long long)ET * HEADS, 256), 256, 0, stream>>>(ei, ASRC, ADST, MENC, DEN, H2, OUT2);
  bias_relu_kernel<<<cdiv((long long)NN * 128, 256), 256, 0, stream>>>(OUT2, b2, (long long)NN * 128, 128);

  // ---------- global mean pool ----------
  hipMemsetAsync(G,   0, (size_t)NG * 128 * 4, stream);
  hipMemsetAsync(CNT, 0, (size_t)NG * 4, stream);
  pool_sum_kernel<<<cdiv((long long)NN * 128, 256), 256, 0, stream>>>(OUT2, batch, G, CNT);
  pool_div_kernel<<<cdiv(NG * 128, 256), 256, 0, stream>>>(G, CNT);

  // ---------- MLP head (WMMA GEMMs, M=64) ----------
  wmma_gemm_f32<<<dim3(2, 4), gblk, 0, stream>>>(G,  fc1_w, fc1_b, G1, 64, 128, 256, 1);
  wmma_gemm_f32<<<dim3(2, 8), gblk, 0, stream>>>(G1, fc2_w, fc2_b, G2, 64, 256, 512, 1);
  wmma_gemm_f32<<<dim3(2, 8), gblk, 0, stream>>>(G2, fc3_w, fc3_b, G3, 64, 512, 512, 1);
  wmma_gemm_f32<<<dim3(2, 2), gblk, 0, stream>>>(G3, fc4_w, fc4_b, (float*)d_out, 64, 512, 100, 0);
}